// EarModel_33792802685581
// MI455X (gfx1250) — compile-verified
//
#include <hip/hip_runtime.h>
#include <cmath>

// ---------------------------------------------------------------------------
// HAAQI ear model for MI455X (gfx1250, wave32).
// Serial IIR chains are blocked: w = particular (16x16 Toeplitz GEMM via
// V_WMMA_F32_16X16X4_F32) + homogeneous (4-state carry, 6000 steps).
// ---------------------------------------------------------------------------

#define NCH   32
#define EARQ  9.26449f
#define MINBW 24.7f
#define TWO_PI 6.283185307179586

typedef __attribute__((ext_vector_type(2))) float v2f;
typedef __attribute__((ext_vector_type(8))) float v8f;

// PAR layout (float indices into workspace param block)
#define P_CF       0
#define P_BW1      32
#define P_AOHCX    64
#define P_BWMINX   96
#define P_KNEEX    128
#define P_CRX      160
#define P_AIHCX    192
#define P_AOHCY    224
#define P_BWMINY   256
#define P_KNEEY    288
#define P_CRY      320
#define P_AIHCY    352
#define P_BWX      384
#define P_BWY      416
#define P_REFAVG   448
#define P_PROCAVG  480
#define P_REFCAVG  512
#define P_PROCCAVG 544
#define P_GDCOR    576   // int[32]
#define P_DELCC    608   // int[32]
#define P_DELB     640   // int[32]
#define P_ALIGN64  672   // u64 (8-byte aligned: 672*4 = 2688)
#define P_CH64     680   // u64[32]
#define PAR_FLOATS 4096

// ------------------------------- device helpers ----------------------------

__device__ __forceinline__ float dev_cf(int i) {
    const float q = EARQ * MINBW;
    int k = 31 - i;
    return -q + expf((float)k * (logf(80.f + q) - logf(8000.f + q)) * (1.f / 31.f)) * (8000.f + q);
}

__device__ __forceinline__ float dev_a(int c, const float* bw) {
    float cfc = dev_cf(c);
    float erb = MINBW + cfc / EARQ;
    float tpt = (float)(TWO_PI / 24000.0);
    return expf(-tpt * bw[c] * erb * 1.019f);
}

__device__ __forceinline__ float gt_gain(float a) {
    float a1 = 4.f * a, a2 = -6.f * a * a, a3 = 4.f * a * a * a, a4 = -a * a * a * a, a5 = 4.f * a * a;
    return 2.f * (1.f - a1 - a2 - a3 - a4) / (1.f + a1 + a5);
}

__device__ __forceinline__ void loss_par(float loss_in, int c, float* ao, float* bwmin,
                                         float* knee, float* cr_out, float* ai) {
    float cr = 1.25f + 2.25f * (float)c / 31.f;
    float mx = 70.f * (1.f - 1.f / cr);
    float thr = 1.25f * mx;
    float loss = fmaxf(loss_in, 0.f);
    float a_ohc, a_ihc;
    if (loss < thr) { a_ohc = 0.8f * loss; a_ihc = 0.2f * loss; }
    else            { a_ohc = 0.8f * thr;  a_ihc = loss - 0.8f * thr; }
    float r = a_ohc / 50.f;
    float r2 = r * r; float r6 = r2 * r2 * r2;
    *ao = a_ohc;
    *bwmin = 1.f + r + 2.f * r6;
    *knee = a_ohc + 30.f;
    float up = 30.f + 70.f / cr;
    *cr_out = (100.f - *knee) / (up + a_ohc - *knee);
    *ai = a_ihc;
}

__device__ float interp_loss(float v, const float* hl, float cf0, float cf31) {
    float fx[8] = { cf0, 250.f, 500.f, 1000.f, 2000.f, 4000.f, 6000.f, cf31 };
    float fy[8] = { hl[0], hl[0], hl[1], hl[2], hl[3], hl[4], hl[5], hl[5] };
    if (v <= fx[0]) return fy[0];
    for (int s = 0; s < 7; ++s) {
        if (v <= fx[s + 1]) {
            float t = (v - fx[s]) / (fx[s + 1] - fx[s]);
            return fy[s] + t * (fy[s + 1] - fy[s]);
        }
    }
    return fy[7];
}

__device__ __forceinline__ unsigned ordkey_signed(float v) {
    unsigned b = __float_as_uint(v);
    return (b & 0x80000000u) ? ~b : (b | 0x80000000u);
}

__device__ __forceinline__ float hash_gauss(unsigned idx, unsigned salt) {
    unsigned s = idx * 747796405u + salt * 2891336453u + 1u;
    unsigned a = s;
    a ^= a >> 17; a *= 0xed5ad4bbu; a ^= a >> 11; a *= 0xac4c1b51u; a ^= a >> 15;
    unsigned b = s + 0x9e3779b9u;
    b ^= b >> 16; b *= 0x7feb352du; b ^= b >> 15; b *= 0x846ca68bu; b ^= b >> 16;
    float u1 = (float)(a >> 8) * (1.0f / 16777216.0f) + 1e-7f;
    float u2 = (float)(b >> 8) * (1.0f / 16777216.0f);
    return sqrtf(-2.f * logf(u1)) * cosf(6.2831853f * u2);
}

// h[k] of 1/(1 - a z^-1)^4 = C(k+3,3) a^k
__device__ __forceinline__ float htap(int d, float a) {
    if (d < 0 || d > 15) return 0.f;
    float binom = (float)((d + 1) * (d + 2) * (d + 3)) * (1.f / 6.f);
    return binom * powf(a, (float)d);
}

// ------------------------------- kernels -----------------------------------

__global__ void k_zero64(unsigned long long* p, int count) {
    int i = blockIdx.x * blockDim.x + threadIdx.x;
    if (i < count) p[i] = 0ull;
}

__global__ void k_params(const float* hl, const int* equalp, float* PAR) {
    int c = threadIdx.x;
    if (c >= NCH) return;
    float cfc = dev_cf(c);
    PAR[P_CF + c] = cfc;
    float cf0 = dev_cf(0), cf31 = dev_cf(31);
    float ao, bm, kn, cr, ai;
    loss_par(100.f, c, &ao, &bm, &kn, &cr, &ai);
    PAR[P_BW1 + c] = bm;
    float lossy = interp_loss(cfc, hl, cf0, cf31);
    loss_par(lossy, c, &ao, &bm, &kn, &cr, &ai);
    PAR[P_AOHCY + c] = ao; PAR[P_BWMINY + c] = bm; PAR[P_KNEEY + c] = kn;
    PAR[P_CRY + c] = cr;   PAR[P_AIHCY + c] = ai;
    float lossx = (*equalp != 0) ? lossy : 0.f;
    loss_par(lossx, c, &ao, &bm, &kn, &cr, &ai);
    PAR[P_AOHCX + c] = ao; PAR[P_BWMINX + c] = bm; PAR[P_KNEEX + c] = kn;
    PAR[P_CRX + c] = cr;   PAR[P_AIHCX + c] = ai;
}

// full-lag cross-correlation + packed argmax of |corr|
__global__ void k_xcorr(const float* __restrict__ x, const float* __restrict__ y,
                        int n, unsigned long long* slot) {
    int m = blockIdx.x * blockDim.x + threadIdx.x;
    int total = 2 * n - 1;
    if (m >= total) return;
    int d = m - (n - 1);
    int t0 = d < 0 ? -d : 0;
    int t1 = d > 0 ? n - d : n;
    float s = 0.f;
    for (int t = t0; t < t1; ++t) s = fmaf(x[t + d], y[t], s);
    unsigned key = __float_as_uint(fabsf(s));
    atomicMax(slot, ((unsigned long long)key << 32) | (unsigned)m);
}

__global__ void k_shift_in(const float* __restrict__ y, float* __restrict__ y2,
                           const unsigned long long* slot, int n) {
    int i = blockIdx.x * blockDim.x + threadIdx.x;
    if (i >= n) return;
    int m = (int)(*slot & 0xffffffffull);
    int delay = (n - 1) - m - 48;   // round(2*24000/1000) = 48
    int idx = i + delay;
    y2[i] = (idx >= 0 && idx < n) ? y[idx] : 0.f;
}

// middle ear: butter1 LP(5k) then butter2 HP(350); 2 serial chains
__global__ void k_mear(const float* x, const float* y2, float* xm, float* ym, int n,
                       float b10, float b11, float a11,
                       float b20, float b21, float b22, float a21, float a22) {
    int tid = threadIdx.x;
    if (tid > 1) return;
    const float* in = tid ? y2 : x;
    float* out = tid ? ym : xm;
    float z = 0.f, z0 = 0.f, z1 = 0.f;
    for (int t = 0; t < n; ++t) {
        float xt = in[t];
        float y1 = b10 * xt + z;   z  = b11 * xt - a11 * y1;
        float yo = b20 * y1 + z0;  z0 = z1 + b21 * y1 - a21 * yo;  z1 = b22 * y1 - a22 * yo;
        out[t] = yo;
    }
}

__device__ __forceinline__ float demod_sample(const float* mid, int t, double w, int ph) {
    double phd = fmod(w * (double)t, TWO_PI);
    float s = (float)phd;
    return mid[t] * (ph == 0 ? cosf(s) : sinf(s));
}

// u[t] = xd[t] + 4a xd[t-1] + 4a^2 xd[t-2] per chain (chain = ph*32 + c)
__global__ void k_demod(const float* __restrict__ mid, const float* __restrict__ bw,
                        float* __restrict__ U, int n) {
    int gid = blockIdx.x * blockDim.x + threadIdx.x;
    int tot = 64 * n;
    if (gid >= tot) return;
    int t = gid % n;
    int chain = gid / n;
    int c = chain & 31, ph = chain >> 5;
    float a = dev_a(c, bw);
    float a1 = 4.f * a, a5 = 4.f * a * a;
    double w = (TWO_PI / 24000.0) * (double)dev_cf(c);
    float d0 = demod_sample(mid, t, w, ph);
    float d1 = (t >= 1) ? demod_sample(mid, t - 1, w, ph) : 0.f;
    float d2 = (t >= 2) ? demod_sample(mid, t - 2, w, ph) : 0.f;
    U[(size_t)chain * n + t] = d0 + a1 * d1 + a5 * d2;
}

// Particular solution of the 4th-order gammatone denominator per 16-sample
// block: P[i,j] = sum_k h[i-k] U[16j+k]  (16x16 Toeplitz GEMM via WMMA f32).
// One wave per (chain, 256-sample tile). Layouts per ISA 7.12.2.
__global__ void k_gt_wmma(const float* __restrict__ U, float* __restrict__ P,
                          const float* __restrict__ bw, int n) {
    int tilesPerChain = n / 256;
    int gb = blockIdx.x;
    int chain = gb / tilesPerChain;
    int tile = gb % tilesPerChain;
    int c = chain & 31;
    int lane = threadIdx.x;
    int i = lane & 15;
    int half = lane >> 4;
    float a = dev_a(c, bw);

    v2f afr[4];
#pragma unroll
    for (int q = 0; q < 4; ++q) {
        int k0 = 4 * q + 2 * half;          // A: VGPR0 holds K=k0, VGPR1 holds K=k0+1
        afr[q].x = htap(i - k0, a);
        afr[q].y = htap(i - k0 - 1, a);
    }

    const float* ub = U + (size_t)chain * n + (size_t)tile * 256;
    int j = i;                              // N index = lane % 16
    v8f acc = { 0, 0, 0, 0, 0, 0, 0, 0 };
#pragma unroll
    for (int q = 0; q < 4; ++q) {
        v2f bfr;                            // B: VGPR0 K=4q+half, VGPR1 K=4q+2+half
        bfr.x = ub[16 * j + 4 * q + half];
        bfr.y = ub[16 * j + 4 * q + 2 + half];
#if defined(__gfx1250__) && __has_builtin(__builtin_amdgcn_wmma_f32_16x16x4_f32)
        acc = __builtin_amdgcn_wmma_f32_16x16x4_f32(false, afr[q], false, bfr,
                                                    (short)0, acc, false, false);
#else
#pragma unroll
        for (int r = 0; r < 8; ++r) acc[r] += afr[q].x * bfr.x + afr[q].y * bfr.y;
#endif
    }
    // C/D: VGPR r holds M = r + 8*half, N = lane%16 -> contiguous 8-float run
    float* pb = P + (size_t)chain * n + (size_t)tile * 256 + 16 * j + 8 * half;
#pragma unroll
    for (int r = 0; r < 8; ++r) pb[r] = acc[r];
}

// Carry: add homogeneous response K(16x4) @ state, 6000 sequential block steps
__global__ void k_gt_carry(float* __restrict__ P, const float* __restrict__ bw, int n) {
    int chain = blockIdx.x * blockDim.x + threadIdx.x;
    if (chain >= 64) return;
    int c = chain & 31;
    float a = dev_a(c, bw);
    float a2 = a * a;
    float c1 = 4.f * a, c2 = -6.f * a2, c3 = 4.f * a2 * a, c4 = -a2 * a2;
    float K[16][4];
#pragma unroll
    for (int m = 0; m < 4; ++m) {
        float p1 = (m == 0) ? 1.f : 0.f, p2 = (m == 1) ? 1.f : 0.f;
        float p3 = (m == 2) ? 1.f : 0.f, p4 = (m == 3) ? 1.f : 0.f;
#pragma unroll
        for (int i = 0; i < 16; ++i) {
            float wi = c1 * p1 + c2 * p2 + c3 * p3 + c4 * p4;
            K[i][m] = wi;
            p4 = p3; p3 = p2; p2 = p1; p1 = wi;
        }
    }
    float* p = P + (size_t)chain * n;
    float s0 = 0.f, s1 = 0.f, s2 = 0.f, s3 = 0.f;
    int nb = n / 16;
    for (int b = 0; b < nb; ++b) {
        float w[16];
#pragma unroll
        for (int i = 0; i < 16; ++i) {
            float pv = p[b * 16 + i];
            w[i] = pv + K[i][0] * s0 + K[i][1] * s1 + K[i][2] * s2 + K[i][3] * s3;
            p[b * 16 + i] = w[i];
        }
        s0 = w[15]; s1 = w[14]; s2 = w[13]; s3 = w[12];
    }
}

__global__ void k_combine(const float* __restrict__ P, float* __restrict__ env,
                          float* __restrict__ bm, const float* __restrict__ bw, int n) {
    int gid = blockIdx.x * blockDim.x + threadIdx.x;
    int tot = NCH * n;
    if (gid >= tot) return;
    int c = gid / n, t = gid % n;
    float a = dev_a(c, bw);
    float g = gt_gain(a);
    float ur = P[(size_t)c * n + t];
    float ui = P[(size_t)(32 + c) * n + t];
    env[gid] = g * sqrtf(ur * ur + ui * ui);
    if (bm) {
        double w = (TWO_PI / 24000.0) * (double)dev_cf(c);
        float ph = (float)fmod(w * (double)t, TWO_PI);
        bm[gid] = g * (ur * cosf(ph) + ui * sinf(ph));
    }
}

// per-channel RMS (+ optional bandwidth_adjust); one block per channel
__global__ void k_reduce(const float* __restrict__ in, int n, float* rms_out,
                         float* bw_out, const float* bwmin, const float* bw1,
                         const int* l1p) {
    __shared__ float sm[256];
    int c = blockIdx.x;
    float s = 0.f;
    for (int t = threadIdx.x; t < n; t += 256) {
        float v = in[(size_t)c * n + t];
        s = fmaf(v, v, s);
    }
    sm[threadIdx.x] = s;
    __syncthreads();
    for (int o = 128; o > 0; o >>= 1) {
        if (threadIdx.x < o) sm[threadIdx.x] += sm[threadIdx.x + o];
        __syncthreads();
    }
    if (threadIdx.x == 0) {
        float rms = sqrtf(sm[0] / (float)n);
        rms_out[c] = rms;
        if (bw_out) {
            float L1 = (float)(*l1p);
            float cdb = 20.f * log10f(rms + 1e-30f) + L1;
            float fr = fminf(fmaxf((cdb - 50.f) / 50.f, 0.f), 1.f);
            bw_out[c] = bwmin[c] + fr * (bw1[c] - bwmin[c]);
        }
    }
}

__global__ void k_cgain(const float* __restrict__ ctrl, float* __restrict__ gn,
                        const float* aohc, const float* knee, const float* cr,
                        const int* l1p, int n) {
    int gid = blockIdx.x * blockDim.x + threadIdx.x;
    int tot = NCH * n;
    if (gid >= tot) return;
    int c = gid / n;
    float L1 = (float)(*l1p);
    float le = L1 + 20.f * log10f(fmaxf(ctrl[gid], 1e-30f));
    le = fminf(fmaxf(le, knee[c]), 100.f);
    gn[gid] = exp10f((-aohc[c] - (le - knee[c]) * (1.f - 1.f / cr[c])) / 20.f);
}

// 800 Hz 1st-order LP smoothing of gain; 32 serial chains
__global__ void k_lpg(float* __restrict__ gn, int n, float b0, float b1, float a1) {
    int c = threadIdx.x;
    if (c >= NCH) return;
    float* g = gn + (size_t)c * n;
    float z = 0.f;
    for (int t = 0; t < n; ++t) {
        float xt = g[t];
        float y = b0 * xt + z;
        z = b1 * xt - a1 * y;
        g[t] = y;
    }
}

__global__ void k_mul(float* __restrict__ a, const float* __restrict__ b, int count) {
    int i = blockIdx.x * blockDim.x + threadIdx.x;
    if (i < count) a[i] *= b[i];
}

__global__ void k_copy(float* __restrict__ d, const float* __restrict__ s, int count) {
    int i = blockIdx.x * blockDim.x + threadIdx.x;
    if (i < count) d[i] = s[i];
}

// per-channel windowed cross-correlation + packed argmax
__global__ void k_ecorr(const float* __restrict__ x, const float* __restrict__ y,
                        int n, int lags, unsigned long long* slots) {
    int gid = blockIdx.x * blockDim.x + threadIdx.x;
    int tot = NCH * 2 * lags;
    if (gid >= tot) return;
    int c = gid / (2 * lags);
    int w = gid % (2 * lags);
    int d = w - lags + 1;
    int t0 = d < 0 ? -d : 0;
    int t1 = d > 0 ? n - d : n;
    const float* xr = x + (size_t)c * n;
    const float* yr = y + (size_t)c * n;
    float s = 0.f;
    for (int t = t0; t < t1; ++t) s = fmaf(xr[t + d], yr[t], s);
    atomicMax(&slots[c], ((unsigned long long)ordkey_signed(s) << 32) | (unsigned)w);
}

__global__ void k_edelay(const unsigned long long* slots, int* delays, int lags) {
    int c = threadIdx.x;
    if (c >= NCH) return;
    int loc = (int)(slots[c] & 0xffffffffull);
    delays[c] = lags - loc - 1;
}

__global__ void k_eshift(const float* __restrict__ y, float* __restrict__ out,
                         const int* delays, int n) {
    int gid = blockIdx.x * blockDim.x + threadIdx.x;
    int tot = NCH * n;
    if (gid >= tot) return;
    int c = gid / n, t = gid % n;
    int idx = t + delays[c];
    out[gid] = (idx >= 0 && idx < n) ? y[(size_t)c * n + idx] : 0.f;
}

__global__ void k_esl(float* __restrict__ env, float* __restrict__ bm,
                      const float* aihc, const int* l1p, int n) {
    int gid = blockIdx.x * blockDim.x + threadIdx.x;
    int tot = NCH * n;
    if (gid >= tot) return;
    int c = gid / n;
    float L1 = (float)(*l1p);
    float e = env[gid];
    float yv = fmaxf(L1 - aihc[c] + 20.f * log10f(e + 1e-30f), 0.f);
    bm[gid] *= yv / (e + 1e-30f);
    env[gid] = yv;
}

// IHC 2-capacitor adaptation; 32 serial chains per signal
__global__ void k_ihc(float* __restrict__ env, float* __restrict__ bm, int n,
                      float r2, float p12, float p23, float a11, float a12,
                      float a21, float a22, float denom, float delta, float gain) {
    int c = threadIdx.x;
    if (c >= NCH) return;
    float* e = env + (size_t)c * n;
    float* b = bm + (size_t)c * n;
    float v1 = 0.f, v2 = 0.f;
    for (int t = 0; t < n; ++t) {
        float v0 = e[t];
        float b1v = v0 * r2 + p12 * v1;
        float b2v = p23 * v2;
        v1 = denom * (a22 * b1v - a12 * b2v);
        v2 = denom * (-a21 * b1v + a11 * b2v);
        float o = gain * fmaxf((v0 - v1) * delta, 0.f);
        b[t] *= (o + 1e-30f) / (v0 + 1e-30f);
        e[t] = o;
    }
}

__global__ void k_noise(float* __restrict__ bm, const int* l1p, unsigned salt, int count) {
    int i = blockIdx.x * blockDim.x + threadIdx.x;
    if (i >= count) return;
    float L1 = (float)(*l1p);
    float ng = exp10f((-10.f - L1) / 20.f);
    bm[i] += ng * hash_gauss((unsigned)i, salt);
}

__global__ void k_gdprep(const float* bwx, int* correct) {
    if (threadIdx.x != 0 || blockIdx.x != 0) return;
    int gdi[NCH];
    int mn = 0x7fffffff;
    for (int c = 0; c < NCH; ++c) {
        float a = dev_a(c, bwx);
        float a2 = a * a;
        float b0 = 1.f, b1 = 4.f * a, b2 = 4.f * a2;
        float d0 = 1.f, d1 = -4.f * a, d2 = 6.f * a2, d3 = -4.f * a2 * a, d4 = a2 * a2;
        float gb = (b1 + 2.f * b2) / (b0 + b1 + b2);
        float ga = (d1 + 2.f * d2 + 3.f * d3 + 4.f * d4) / (d0 + d1 + d2 + d3 + d4);
        float gd = gb - ga;
        gdi[c] = (int)rintf(gd);
        if (gdi[c] < mn) mn = gdi[c];
    }
    int mx = -0x7fffffff;
    for (int c = 0; c < NCH; ++c) { gdi[c] -= mn; if (gdi[c] > mx) mx = gdi[c]; }
    for (int c = 0; c < NCH; ++c) correct[c] = mx - gdi[c];
}

__global__ void k_gdshift(const float* __restrict__ in, float* __restrict__ out,
                          const int* correct, int n) {
    int gid = blockIdx.x * blockDim.x + threadIdx.x;
    int tot = NCH * n;
    if (gid >= tot) return;
    int c = gid / n, t = gid % n;
    int idx = t - correct[c];
    out[gid] = (idx >= 0) ? in[(size_t)c * n + idx] : 0.f;
}

__global__ void k_avesl(const float* PAR, const int* l1p,
                        float* ref_sl, float* proc_sl, float* fs_out) {
    int c = threadIdx.x;
    if (c == 0 && fs_out) *fs_out = 24000.f;
    if (c >= NCH) return;
    float L1 = (float)(*l1p);
    // x side
    {
        float cdb = L1 + 20.f * log10f(fmaxf(PAR[P_REFCAVG + c], 1e-30f));
        cdb = fminf(fmaxf(cdb, PAR[P_KNEEX + c]), 100.f);
        float g = -PAR[P_AOHCX + c] - (cdb - PAR[P_KNEEX + c]) * (1.f - 1.f / PAR[P_CRX + c]);
        float sl = L1 + 20.f * log10f(fmaxf(PAR[P_REFAVG + c], 1e-30f)) + g - PAR[P_AIHCX + c];
        ref_sl[c] = fmaxf(sl, 0.f);
    }
    // y side
    {
        float cdb = L1 + 20.f * log10f(fmaxf(PAR[P_PROCCAVG + c], 1e-30f));
        cdb = fminf(fmaxf(cdb, PAR[P_KNEEY + c]), 100.f);
        float g = -PAR[P_AOHCY + c] - (cdb - PAR[P_KNEEY + c]) * (1.f - 1.f / PAR[P_CRY + c]);
        float sl = L1 + 20.f * log10f(fmaxf(PAR[P_PROCAVG + c], 1e-30f)) + g - PAR[P_AIHCY + c];
        proc_sl[c] = fmaxf(sl, 0.f);
    }
}

// ------------------------------- host driver -------------------------------

extern "C" void kernel_launch(void* const* d_in, const int* in_sizes, int n_in,
                              void* d_out, int out_size, void* d_ws, size_t ws_size,
                              hipStream_t stream) {
    const float* ref  = (const float*)d_in[0];
    const float* proc = (const float*)d_in[2];
    const float* hl   = (const float*)d_in[4];
    const int* equalp = (const int*)d_in[5];
    const int* l1p    = (const int*)d_in[6];

    const int n = in_sizes[0];              // 96000
    const size_t Cn = (size_t)NCH * n;
    const int iCn = (int)Cn;
    const int lags = (2400 < n) ? 2400 : n;

    float* W = (float*)d_ws;
    float* PAR  = W;
    float* Y24  = W + PAR_FLOATS;
    float* XMID = Y24 + n;
    float* YMID = XMID + n;
    float* XC   = YMID + n;
    float* YC   = XC + Cn;
    float* XE   = YC + Cn;     // xenv -> xcc -> xdb
    float* XB   = XE + Cn;     // xbm  -> xb  -> xbm_n
    float* YE   = XB + Cn;
    float* YB   = YE + Cn;
    float* UB   = YB + Cn;     // 2*Cn demod inputs
    float* PB   = UB + 2 * Cn; // 2*Cn particular/filtered
    float* GN   = PB + 2 * Cn; // Cn gain
    (void)ws_size; (void)n_in; (void)out_size;

    int* gdcor   = (int*)(PAR + P_GDCOR);
    int* delcc   = (int*)(PAR + P_DELCC);
    int* delb    = (int*)(PAR + P_DELB);
    unsigned long long* alignSlot = (unsigned long long*)(PAR + P_ALIGN64);
    unsigned long long* chSlots   = (unsigned long long*)(PAR + P_CH64);

    float* out = (float*)d_out;
    float* o_xdb  = out;
    float* o_xbm  = out + Cn;
    float* o_ydb  = out + 2 * Cn;
    float* o_ybm  = out + 3 * Cn;
    float* o_rsl  = out + 4 * Cn;
    float* o_psl  = o_rsl + NCH;
    float* o_fs   = o_psl + NCH;

    // host-side filter constants
    const double SQ2 = 1.4142135623730951;
    double c1d = std::tan(M_PI * 5000.0 / 24000.0);
    float b10 = (float)(c1d / (1 + c1d)), b11 = b10, a11 = (float)((c1d - 1) / (1 + c1d));
    double c2d = std::tan(M_PI * 350.0 / 24000.0);
    double dd = 1 + SQ2 * c2d + c2d * c2d;
    float b20 = (float)(1.0 / dd), b21 = (float)(-2.0 / dd), b22 = (float)(1.0 / dd);
    float a21 = (float)(2 * (c2d * c2d - 1) / dd);
    float a22 = (float)((1 - SQ2 * c2d + c2d * c2d) / dd);
    double c8d = std::tan(M_PI * 800.0 / 24000.0);
    float g_b0 = (float)(c8d / (1 + c8d)), g_b1 = g_b0, g_a1 = (float)((c8d - 1) / (1 + c8d));
    // IHC constants (delta = 2)
    double r1 = 0.5, r2 = 0.25, r3 = 0.25, T = 1.0 / 24000.0;
    double cc1 = 0.002 * (r1 + r2) / (r1 * r2), cc2 = 0.06 / (r2 + r3);
    double A11 = r1 + r2 + r1 * r2 * (cc1 / T), A12 = -r1, A21 = -r3;
    double A22 = r2 + r3 + r2 * r3 * (cc2 / T);
    double DEN = 1.0 / (A11 * A22 - A21 * A12);
    double P12 = r1 * r2 * (cc1 / T), P23 = r2 * r3 * (cc2 / T);
    float ihc_gain = (float)(1.0 / (1.0 + 1e-5));

    const int TB = 256;
    dim3 blk(TB);
    auto g1 = [&](int count) { return dim3((count + TB - 1) / TB); };
    int tiles = n / 256;

    // ---- input_align ----
    k_zero64<<<1, 64, 0, stream>>>(alignSlot, 1);
    k_xcorr<<<g1(2 * n - 1), blk, 0, stream>>>(ref, proc, n, alignSlot);
    k_shift_in<<<g1(n), blk, 0, stream>>>(proc, Y24, alignSlot, n);

    // ---- middle ear + params ----
    k_mear<<<1, 32, 0, stream>>>(ref, Y24, XMID, YMID, n, b10, b11, a11, b20, b21, b22, a21, a22);
    k_params<<<1, 32, 0, stream>>>(hl, equalp, PAR);

    // ---- gammatone control pass (bw_1) ----
    for (int s = 0; s < 2; ++s) {
        const float* mid = s ? YMID : XMID;
        float* env = s ? YC : XC;
        k_demod<<<g1(64 * n), blk, 0, stream>>>(mid, PAR + P_BW1, UB, n);
        k_gt_wmma<<<dim3(64 * tiles), dim3(32), 0, stream>>>(UB, PB, PAR + P_BW1, n);
        k_gt_carry<<<2, 32, 0, stream>>>(PB, PAR + P_BW1, n);
        k_combine<<<g1(iCn), blk, 0, stream>>>(PB, env, nullptr, PAR + P_BW1, n);
    }

    // ---- bandwidth_adjust (also stores control RMS for ave_sl) ----
    k_reduce<<<32, 256, 0, stream>>>(XC, n, PAR + P_REFCAVG, PAR + P_BWX,
                                     PAR + P_BWMINX, PAR + P_BW1, l1p);
    k_reduce<<<32, 256, 0, stream>>>(YC, n, PAR + P_PROCCAVG, PAR + P_BWY,
                                     PAR + P_BWMINY, PAR + P_BW1, l1p);

    // ---- gammatone pass 2 (adjusted bandwidths), env + bm ----
    for (int s = 0; s < 2; ++s) {
        const float* mid = s ? YMID : XMID;
        float* env = s ? YE : XE;
        float* bm  = s ? YB : XB;
        const float* bwp = s ? (PAR + P_BWY) : (PAR + P_BWX);
        k_demod<<<g1(64 * n), blk, 0, stream>>>(mid, bwp, UB, n);
        k_gt_wmma<<<dim3(64 * tiles), dim3(32), 0, stream>>>(UB, PB, bwp, n);
        k_gt_carry<<<2, 32, 0, stream>>>(PB, bwp, n);
        k_combine<<<g1(iCn), blk, 0, stream>>>(PB, env, bm, bwp, n);
    }

    // ---- envelope RMS (pre-compression) for ave_sl ----
    k_reduce<<<32, 256, 0, stream>>>(XE, n, PAR + P_REFAVG, nullptr, nullptr, nullptr, l1p);
    k_reduce<<<32, 256, 0, stream>>>(YE, n, PAR + P_PROCAVG, nullptr, nullptr, nullptr, l1p);

    // ---- env_compress_bm ----
    k_cgain<<<g1(iCn), blk, 0, stream>>>(XC, GN, PAR + P_AOHCX, PAR + P_KNEEX, PAR + P_CRX, l1p, n);
    k_lpg<<<1, 32, 0, stream>>>(GN, n, g_b0, g_b1, g_a1);
    k_mul<<<g1(iCn), blk, 0, stream>>>(XE, GN, iCn);
    k_mul<<<g1(iCn), blk, 0, stream>>>(XB, GN, iCn);
    k_cgain<<<g1(iCn), blk, 0, stream>>>(YC, GN, PAR + P_AOHCY, PAR + P_KNEEY, PAR + P_CRY, l1p, n);
    k_lpg<<<1, 32, 0, stream>>>(GN, n, g_b0, g_b1, g_a1);
    k_mul<<<g1(iCn), blk, 0, stream>>>(YE, GN, iCn);
    k_mul<<<g1(iCn), blk, 0, stream>>>(YB, GN, iCn);

    // ---- envelope_align (cc pair, then bm pair) ----
    k_zero64<<<1, 64, 0, stream>>>(chSlots, NCH);
    k_ecorr<<<g1(NCH * 2 * lags), blk, 0, stream>>>(XE, YE, n, lags, chSlots);
    k_edelay<<<1, 32, 0, stream>>>(chSlots, delcc, lags);
    k_eshift<<<g1(iCn), blk, 0, stream>>>(YE, PB, delcc, n);
    k_copy<<<g1(iCn), blk, 0, stream>>>(YE, PB, iCn);

    k_zero64<<<1, 64, 0, stream>>>(chSlots, NCH);
    k_ecorr<<<g1(NCH * 2 * lags), blk, 0, stream>>>(XB, YB, n, lags, chSlots);
    k_edelay<<<1, 32, 0, stream>>>(chSlots, delb, lags);
    k_eshift<<<g1(iCn), blk, 0, stream>>>(YB, PB, delb, n);
    k_copy<<<g1(iCn), blk, 0, stream>>>(YB, PB, iCn);

    // ---- envelope_sl ----
    k_esl<<<g1(iCn), blk, 0, stream>>>(XE, XB, PAR + P_AIHCX, l1p, n);
    k_esl<<<g1(iCn), blk, 0, stream>>>(YE, YB, PAR + P_AIHCY, l1p, n);

    // ---- ihc_adapt ----
    k_ihc<<<1, 32, 0, stream>>>(XE, XB, n, (float)r2, (float)P12, (float)P23,
                                (float)A11, (float)A12, (float)A21, (float)A22,
                                (float)DEN, 2.0f, ihc_gain);
    k_ihc<<<1, 32, 0, stream>>>(YE, YB, n, (float)r2, (float)P12, (float)P23,
                                (float)A11, (float)A12, (float)A21, (float)A22,
                                (float)DEN, 2.0f, ihc_gain);

    // ---- additive IHC noise ----
    k_noise<<<g1(iCn), blk, 0, stream>>>(XB, l1p, 0x1234567u, iCn);
    k_noise<<<g1(iCn), blk, 0, stream>>>(YB, l1p, 0x89abcdfu, iCn);

    // ---- group delay compensation (bw_x for all four) -> d_out ----
    k_gdprep<<<1, 1, 0, stream>>>(PAR + P_BWX, gdcor);
    k_gdshift<<<g1(iCn), blk, 0, stream>>>(XE, o_xdb, gdcor, n);
    k_gdshift<<<g1(iCn), blk, 0, stream>>>(XB, o_xbm, gdcor, n);
    k_gdshift<<<g1(iCn), blk, 0, stream>>>(YE, o_ydb, gdcor, n);
    k_gdshift<<<g1(iCn), blk, 0, stream>>>(YB, o_ybm, gdcor, n);

    // ---- average sensation levels + FS ----
    k_avesl<<<1, 32, 0, stream>>>(PAR, l1p, o_rsl, o_psl, o_fs);
}